// GraphSAGE_33724083208194
// MI455X (gfx1250) — compile-verified
//
#include <hip/hip_runtime.h>
#include <hip/hip_bf16.h>

// GraphSAGE (2-layer, mean aggregation) for MI455X / gfx1250.
//
// Roofline reasoning: the scatter-mean over 1.6M edges x 128 feats dominates
// (410M atomic f32 adds, ~1.6GB gather traffic). The 51MB feature matrix fits
// in the 192MB L2, so gathers/atomics are L2-resident. The four 100k x 128 x 128
// GEMMs are only ~13 GFLOP and are memory-bound (~150MB/layer @ 23.3TB/s), so
// fp32 precision is free: we use V_WMMA_F32_16X16X4_F32 (native fp32 WMMA)
// instead of downcasting.

typedef __attribute__((ext_vector_type(2))) float v2f;
typedef __attribute__((ext_vector_type(8))) float v8f;

#define DFEAT 128

// ---------------------------------------------------------------------------
// Degree accumulation: one thread per edge, float atomic add into deg[dst].
// ---------------------------------------------------------------------------
__global__ void deg_kernel(const int* __restrict__ dst, float* __restrict__ deg,
                           int nEdges) {
    int e = blockIdx.x * blockDim.x + threadIdx.x;
    if (e < nEdges) {
        atomicAdd(&deg[dst[e]], 1.0f);
    }
}

__global__ void deg_inv_kernel(float* __restrict__ deg, int n) {
    int i = blockIdx.x * blockDim.x + threadIdx.x;
    if (i < n) {
        deg[i] = 1.0f / fmaxf(deg[i], 1.0f);
    }
}

// ---------------------------------------------------------------------------
// Edge scatter: one wave (32 lanes) per edge; lane L moves 4 contiguous floats
// [4L..4L+3] of the 128-float source row -> coalesced 512B read per wave,
// then 4 non-returning global_atomic_add_f32 (STOREcnt path, L2-resident).
// ---------------------------------------------------------------------------
__global__ void scatter_add_kernel(const float* __restrict__ x,
                                   const int* __restrict__ src,
                                   const int* __restrict__ dst,
                                   float* __restrict__ agg, int nEdges) {
    int idx = blockIdx.x * blockDim.x + threadIdx.x;
    int e = idx >> 5;  // 32 lanes per edge (wave32)
    if (e >= nEdges) return;
    int off = (idx & 31) * 4;
    int s = src[e];
    int d = dst[e];
    const float4 v = *(const float4*)&x[(size_t)s * DFEAT + off];
    float* p = &agg[(size_t)d * DFEAT + off];
    atomicAdd(p + 0, v.x);
    atomicAdd(p + 1, v.y);
    atomicAdd(p + 2, v.z);
    atomicAdd(p + 3, v.w);
}

// ---------------------------------------------------------------------------
// Fused SAGE GEMM:  out[n,:] = X[n,:] @ Wself^T + (AGG[n,:]*deg_inv[n]) @ Wneigh^T + b
// Block = 16 node rows x 128 output cols; 8 waves, each wave owns one 16x16
// output tile and runs 32 K-steps x 2 matrices = 64 v_wmma_f32_16x16x4_f32.
//
// Fragment layouts (ISA 7.12.2, 32-bit):
//   A (16x4):  lane<16: row=lane,   holds K=k0,k0+1 ; lane>=16: row=lane-16, K=k0+2,k0+3
//   B (4x16):  lane<16: col=lane,   holds K=k0,k0+1 ; lane>=16: col=lane-16, K=k0+2,k0+3
//   C (16x16): VGPR r: lane<16 -> (M=r,  N=lane) ; lane>=16 -> (M=r+8, N=lane-16)
// ---------------------------------------------------------------------------
__global__ void sage_gemm_kernel(const float* __restrict__ X,
                                 const float* __restrict__ AGG,
                                 const float* __restrict__ deg_inv,
                                 const float* __restrict__ Wself,
                                 const float* __restrict__ Wneigh,
                                 const float* __restrict__ bias,
                                 float* __restrict__ out,
                                 int nNodes, int doRelu) {
    // LDS staging, stride 132 (=128+4) so column reads hit distinct banks.
    __shared__ float sX[16][132];
    __shared__ float sH[16][132];

    const int row0 = blockIdx.x * 16;
    const int tid  = threadIdx.x;

    // Cooperative load: 16 rows x 32 float4 per array; 512 float4 / 256 thr.
    // Uniform trip count -> no EXEC divergence before the WMMAs.
    for (int i = tid; i < 16 * 32; i += 256) {
        int r  = i >> 5;
        int c4 = (i & 31) * 4;
        int gr = row0 + r;
        if (gr >= nNodes) gr = nNodes - 1;  // clamp (N=100000 is exact anyway)
        const float4 xv = *(const float4*)&X[(size_t)gr * DFEAT + c4];
        const float4 av = *(const float4*)&AGG[(size_t)gr * DFEAT + c4];
        const float  di = deg_inv[gr];
        sX[r][c4 + 0] = xv.x;  sX[r][c4 + 1] = xv.y;
        sX[r][c4 + 2] = xv.z;  sX[r][c4 + 3] = xv.w;
        sH[r][c4 + 0] = av.x * di;  sH[r][c4 + 1] = av.y * di;
        sH[r][c4 + 2] = av.z * di;  sH[r][c4 + 3] = av.w * di;
    }
    __syncthreads();

    const int lane = tid & 31;
    const int wave = tid >> 5;          // 0..7 -> 16-col output tile
    const int colBase = wave * 16;
    const int half = lane >> 4;         // 0 or 1 (K-offset half of the wave)
    const int l16  = lane & 15;

    v8f c = {0.f, 0.f, 0.f, 0.f, 0.f, 0.f, 0.f, 0.f};

    // W row pointers for this lane's output column (B[k][j] = W[j][k]).
    const float* __restrict__ ws = Wself  + (size_t)(colBase + l16) * DFEAT;
    const float* __restrict__ wn = Wneigh + (size_t)(colBase + l16) * DFEAT;

#pragma unroll
    for (int k0 = 0; k0 < DFEAT; k0 += 4) {
        const int ka = k0 + half * 2;

        v2f a, b;
        // Self path: A from LDS X tile, B from Wself (hot in L2, 64KB).
        a.x = sX[l16][ka];  a.y = sX[l16][ka + 1];
        b.x = ws[ka];       b.y = ws[ka + 1];
        c = __builtin_amdgcn_wmma_f32_16x16x4_f32(
                /*neg_a=*/false, a, /*neg_b=*/false, b,
                /*c_mod=*/(short)0, c, /*reuse_a=*/false, /*reuse_b=*/false);

        // Neighbor path: A from deg-scaled aggregate tile, B from Wneigh.
        a.x = sH[l16][ka];  a.y = sH[l16][ka + 1];
        b.x = wn[ka];       b.y = wn[ka + 1];
        c = __builtin_amdgcn_wmma_f32_16x16x4_f32(
                false, a, false, b, (short)0, c, false, false);
    }

    // Epilogue: bias + optional ReLU, store C tile.
    const int col = colBase + l16;
    const float bv = bias[col];
#pragma unroll
    for (int r = 0; r < 8; ++r) {
        const int gr = row0 + r + half * 8;
        if (gr < nNodes) {
            float v = c[r] + bv;
            if (doRelu) v = fmaxf(v, 0.0f);
            out[(size_t)gr * DFEAT + col] = v;
        }
    }
}

// ---------------------------------------------------------------------------
// Launcher. Inputs (setup_inputs order):
//   0 feat [N*D] f32, 1 src [E] i32, 2 dst [E] i32,
//   3 W_self1 [D*D], 4 W_neigh1 [D*D], 5 b1 [D],
//   6 W_self2 [D*D], 7 W_neigh2 [D*D], 8 b2 [D]
// Workspace: deg[N] | agg[N*D] | h[N*D]  (~103 MB)
// ---------------------------------------------------------------------------
extern "C" void kernel_launch(void* const* d_in, const int* in_sizes, int n_in,
                              void* d_out, int out_size, void* d_ws, size_t ws_size,
                              hipStream_t stream) {
    const float* feat = (const float*)d_in[0];
    const int*   src  = (const int*)d_in[1];
    const int*   dst  = (const int*)d_in[2];
    const float* Ws1  = (const float*)d_in[3];
    const float* Wn1  = (const float*)d_in[4];
    const float* b1   = (const float*)d_in[5];
    const float* Ws2  = (const float*)d_in[6];
    const float* Wn2  = (const float*)d_in[7];
    const float* b2   = (const float*)d_in[8];
    float* out = (float*)d_out;

    const int N = in_sizes[0] / DFEAT;  // 100000
    const int E = in_sizes[1];          // 1600000

    float* deg = (float*)d_ws;
    float* agg = deg + N;
    float* h   = agg + (size_t)N * DFEAT;

    // Zero degree + aggregate buffers (capture-safe async memset).
    hipMemsetAsync(d_ws, 0, ((size_t)N * DFEAT + (size_t)N) * sizeof(float), stream);

    // Degrees -> reciprocal (stored in place in deg).
    deg_kernel<<<(E + 255) / 256, 256, 0, stream>>>(dst, deg, E);
    deg_inv_kernel<<<(N + 255) / 256, 256, 0, stream>>>(deg, N);

    const int scatterBlocks = (int)(((long long)E * 32 + 255) / 256);
    const int gemmBlocks    = (N + 15) / 16;

    // Layer 1
    scatter_add_kernel<<<scatterBlocks, 256, 0, stream>>>(feat, src, dst, agg, E);
    sage_gemm_kernel<<<gemmBlocks, 256, 0, stream>>>(feat, agg, deg, Ws1, Wn1, b1,
                                                     h, N, /*relu=*/1);

    // Re-zero aggregate, Layer 2
    hipMemsetAsync(agg, 0, (size_t)N * DFEAT * sizeof(float), stream);
    scatter_add_kernel<<<scatterBlocks, 256, 0, stream>>>(h, src, dst, agg, E);
    sage_gemm_kernel<<<gemmBlocks, 256, 0, stream>>>(h, agg, deg, Ws2, Wn2, b2,
                                                     out, N, /*relu=*/0);
}